// CondBlock_15135464751283
// MI455X (gfx1250) — compile-verified
//
#include <hip/hip_runtime.h>

#define B_   16
#define P_   12
#define N_   2048
#define H_   64
#define E_   16384
#define NH_  (N_ * H_)          // 131072 = 2^17
#define EPS_ 1e-5f

typedef float v2f __attribute__((ext_vector_type(2)));
typedef float v8f __attribute__((ext_vector_type(8)));

// ---------------------------------------------------------------------------
// Degree count: deg[dst] += 1 per edge (deg pre-zeroed by memset).
// ---------------------------------------------------------------------------
__global__ void deg_kernel(const long long* __restrict__ ei, float* __restrict__ deg) {
    int e = blockIdx.x * 256 + threadIdx.x;
    if (e < E_) {
        int dst = (int)ei[E_ + e];
        unsafeAtomicAdd(&deg[dst], 1.0f);
    }
}

// ---------------------------------------------------------------------------
// Per-edge symmetric norm: dinv[src]*dinv[dst].
// ---------------------------------------------------------------------------
__global__ void edgenorm_kernel(const long long* __restrict__ ei,
                                const float* __restrict__ deg,
                                float* __restrict__ nrm) {
    int e = blockIdx.x * 256 + threadIdx.x;
    if (e < E_) {
        int s = (int)ei[e];
        int d = (int)ei[E_ + e];
        float ds = deg[s], dd = deg[d];
        float is = ds > 0.f ? rsqrtf(fmaxf(ds, 1.f)) : 0.f;
        float id = dd > 0.f ? rsqrtf(fmaxf(dd, 1.f)) : 0.f;
        nrm[e] = is * id;
    }
}

// ---------------------------------------------------------------------------
// Per-(b,p) mean / rstd over NH_ elements. One block per (b,p).
// ---------------------------------------------------------------------------
__global__ void stats_kernel(const float* __restrict__ in, float* __restrict__ stats) {
    __shared__ float ssum[256];
    __shared__ float ssq[256];
    int bp = blockIdx.x;
    const float4* p = (const float4*)(in + (size_t)bp * NH_);
    float s = 0.f, q = 0.f;
    for (int i = threadIdx.x; i < NH_ / 4; i += 256) {
        float4 v = p[i];
        s += v.x + v.y + v.z + v.w;
        q += v.x * v.x + v.y * v.y + v.z * v.z + v.w * v.w;
    }
    ssum[threadIdx.x] = s;
    ssq[threadIdx.x]  = q;
    __syncthreads();
    for (int off = 128; off > 0; off >>= 1) {
        if ((int)threadIdx.x < off) {
            ssum[threadIdx.x] += ssum[threadIdx.x + off];
            ssq[threadIdx.x]  += ssq[threadIdx.x + off];
        }
        __syncthreads();
    }
    if (threadIdx.x == 0) {
        float inv  = 1.0f / (float)NH_;
        float mean = ssum[0] * inv;
        float var  = ssq[0] * inv - mean * mean;
        stats[bp * 2 + 0] = mean;
        stats[bp * 2 + 1] = rsqrtf(var + EPS_);
    }
}

// ---------------------------------------------------------------------------
// Fused: norm1(x) gathered at src, scaled by edge norm, scatter-add to y[dst].
// One thread per (bp, e, h). Total = 192 * 16384 * 64 = 201326592 (exact grid).
// ---------------------------------------------------------------------------
__global__ void scatter_kernel(const float* __restrict__ x,
                               const long long* __restrict__ ei,
                               const float* __restrict__ gw,
                               const float* __restrict__ gb,
                               const float* __restrict__ stats1,
                               const float* __restrict__ nrm,
                               float* __restrict__ y) {
    unsigned idx = blockIdx.x * 256u + threadIdx.x;
    int bp       = idx >> 20;            // / (E_*H_) = 2^20
    unsigned rem = idx & 0xFFFFFu;
    int e = rem >> 6;                    // / H_
    int h = rem & 63;
    int s = (int)ei[e];
    int d = (int)ei[E_ + e];
    float m = stats1[bp * 2 + 0];
    float r = stats1[bp * 2 + 1];
    int so = s * H_ + h;
    float xv = x[(size_t)bp * NH_ + so];
    float hv = (xv - m) * r * gw[so] + gb[so];
    unsafeAtomicAdd(&y[(size_t)bp * NH_ + d * H_ + h], hv * nrm[e]);
}

// ---------------------------------------------------------------------------
// Fused norm2 + 12x12 channel-mix GEMM via V_WMMA_F32_16X16X4_F32.
// One wave = one 16(q, padded from 12) x 16(column) output tile.
// K-loop over p in chunks of 4.  Total tiles = B_*NH_/16 = 131072; 4 waves/blk.
// A fragments hoisted + branchless (no exec-mask churn between WMMAs).
// y read once -> non-temporal loads; out written once -> non-temporal stores.
// ---------------------------------------------------------------------------
__global__ void out_wmma_kernel(const float* __restrict__ y,
                                const float* __restrict__ stats2,
                                const float* __restrict__ tw,
                                const float* __restrict__ tb,
                                const float* __restrict__ cw,
                                const float* __restrict__ cb,
                                float* __restrict__ out) {
    const int lane = threadIdx.x & 31;
    const int wave = threadIdx.x >> 5;
    const int tile = blockIdx.x * 4 + wave;
    const int col0 = tile * 16;
    const int b    = col0 >> 17;                 // / NH_
    const int nh   = (col0 & (NH_ - 1)) + (lane & 15);
    const bool hi  = lane >= 16;
    const int qrow = lane & 15;

    // Branchless zero-padding for q rows 12..15: clamp address, mask value.
    const int   qc = (qrow < P_) ? qrow : 0;
    const float qm = (qrow < P_) ? 1.f : 0.f;
    const int   pa = hi ? 2 : 0;   // A layout: lo lanes K{k0,k0+1}, hi lanes K{k0+2,k0+3}

    v2f afrag[3];
    #pragma unroll
    for (int k = 0; k < 3; ++k) {
        afrag[k].x = cw[qc * P_ + 4 * k + pa + 0] * qm;
        afrag[k].y = cw[qc * P_ + 4 * k + pa + 1] * qm;
    }

    const float twv = tw[nh];
    const float tbv = tb[nh];

    v8f acc = {};
    #pragma unroll
    for (int k = 0; k < 3; ++k) {
        const int p0  = 4 * k + pa;
        const int bp0 = b * P_ + p0;
        float m0 = stats2[bp0 * 2 + 0], r0 = stats2[bp0 * 2 + 1];
        float m1 = stats2[bp0 * 2 + 2], r1 = stats2[bp0 * 2 + 3];
        float y0 = __builtin_nontemporal_load(&y[(size_t)bp0 * NH_ + nh]);
        float y1 = __builtin_nontemporal_load(&y[(size_t)(bp0 + 1) * NH_ + nh]);
        v2f bf;
        bf.x = (y0 - m0) * r0 * twv + tbv;
        bf.y = (y1 - m1) * r1 * twv + tbv;

        acc = __builtin_amdgcn_wmma_f32_16x16x4_f32(
            /*neg_a=*/false, afrag[k], /*neg_b=*/false, bf,
            /*c_mod=*/(short)0, acc, /*reuse_a=*/false, /*reuse_b=*/false);
    }

    // Store rows q < 12 with bias (divergence only after all WMMAs).
    #pragma unroll
    for (int r = 0; r < 8; ++r) {
        int q = r + (hi ? 8 : 0);
        if (q < P_) {
            __builtin_nontemporal_store(acc[r] + cb[q],
                                        &out[(size_t)(b * P_ + q) * NH_ + nh]);
        }
    }
}

// ---------------------------------------------------------------------------
extern "C" void kernel_launch(void* const* d_in, const int* in_sizes, int n_in,
                              void* d_out, int out_size, void* d_ws, size_t ws_size,
                              hipStream_t stream) {
    const float*     x  = (const float*)d_in[0];
    const long long* ei = (const long long*)d_in[1];   // int64 edge_index [2,E]
    const float*     gw = (const float*)d_in[2];
    const float*     gb = (const float*)d_in[3];
    const float*     tw = (const float*)d_in[4];
    const float*     tb = (const float*)d_in[5];
    const float*     cw = (const float*)d_in[6];
    const float*     cb = (const float*)d_in[7];
    float* out = (float*)d_out;

    float* ws     = (float*)d_ws;
    float* y      = ws;                                  // [B,P,N,H] accumulator
    float* stats1 = ws + (size_t)B_ * P_ * NH_;          // [B*P][mean,rstd]
    float* stats2 = stats1 + 2 * B_ * P_;
    float* deg    = stats2 + 2 * B_ * P_;                // [N]
    float* nrm    = deg + N_;                            // [E]

    // Zero y accumulator + stats + deg every call (deterministic replay).
    size_t zero_bytes = ((size_t)B_ * P_ * NH_ + 4 * B_ * P_ + N_) * sizeof(float);
    hipMemsetAsync(d_ws, 0, zero_bytes, stream);

    deg_kernel<<<E_ / 256, 256, 0, stream>>>(ei, deg);
    edgenorm_kernel<<<E_ / 256, 256, 0, stream>>>(ei, deg, nrm);
    stats_kernel<<<B_ * P_, 256, 0, stream>>>(x, stats1);
    scatter_kernel<<<(B_ * P_ * E_ * H_) / 256, 256, 0, stream>>>(x, ei, gw, gb, stats1, nrm, y);
    stats_kernel<<<B_ * P_, 256, 0, stream>>>(y, stats2);
    out_wmma_kernel<<<(B_ * NH_ / 16) / 4, 128, 0, stream>>>(y, stats2, tw, tb, cw, cb, out);
}